// SECommEncoder_52853867544720
// MI455X (gfx1250) — compile-verified
//
#include <hip/hip_runtime.h>
#include <hip/hip_bf16.h>

// ---------------------------------------------------------------------------
// GraphConv x2 (DGL norm='both', self-loops) for MI455X (gfx1250, wave32).
// Dense GEMMs use V_WMMA_F32_16X16X4_F32 (exact fp32 on the matrix pipe).
// SpMM uses hardware float atomics (global_atomic_add_f32).
// ---------------------------------------------------------------------------

typedef __attribute__((ext_vector_type(2))) float v2f;
typedef __attribute__((ext_vector_type(8))) float v8f;

#define IN_CH  256
#define HID_CH 256
#define OUT_CH 128

// ---------------------------------------------------------------------------
// Degree counting: integer atomics into count buffers.
// ---------------------------------------------------------------------------
__global__ void degree_kernel(const int* __restrict__ src, const int* __restrict__ dst,
                              int* __restrict__ cnt_src, int* __restrict__ cnt_dst, int E) {
    int i = blockIdx.x * blockDim.x + threadIdx.x;
    if (i < E) {
        atomicAdd(&cnt_src[src[i]], 1);
        atomicAdd(&cnt_dst[dst[i]], 1);
    }
}

// counts (int) -> rsqrt(count + 1) (float), in place. Self-loop adds the +1,
// and guarantees deg >= 1 so max(deg,1) is implicit.
__global__ void norm_kernel(float* __restrict__ buf, int n) {
    int i = blockIdx.x * blockDim.x + threadIdx.x;
    if (i < n) {
        int c = ((const int*)buf)[i];
        buf[i] = rsqrtf((float)(c + 1));
    }
}

// ---------------------------------------------------------------------------
// H = (X @ W) * norm_src[row], fp32 WMMA 16x16x4.
// Block: 16 rows of X staged in LDS; one wave per 16-column tile of W.
// blockDim.x == (NDIM/16)*32.  gridDim.x == N/16 (N divisible by 16).
// Per the gfx1250 ISA VGPR layouts:
//   A 16x4 : lanes 0-15 -> M=lane, VGPR0=K0,VGPR1=K1 ; lanes 16-31 -> K2,K3
//   B 4x16 : VGPR0 -> K0 (lanes 0-15) / K2 (lanes 16-31), VGPR1 -> K1 / K3
//   C/D    : VGPR v -> M=v (lanes 0-15) / M=v+8 (lanes 16-31), N=lane&15
// ---------------------------------------------------------------------------
template <int KDIM, int NDIM>
__global__ void gemm_norm_kernel(const float* __restrict__ X, const float* __restrict__ W,
                                 const float* __restrict__ norm, float* __restrict__ H) {
    __shared__ float As[16][KDIM + 1];   // +1 pad: 16 rows hit distinct LDS banks

    const int tid  = threadIdx.x;
    const int row0 = blockIdx.x * 16;

    // Coalesced cooperative load of the 16 x KDIM A-tile.
    for (int i = tid; i < 16 * KDIM; i += blockDim.x) {
        int r = i / KDIM;
        int c = i - r * KDIM;
        As[r][c] = X[(size_t)(row0 + r) * KDIM + c];
    }
    __syncthreads();

    const int lane    = tid & 31;
    const int wave    = tid >> 5;
    const int half    = lane >> 4;   // 0: lanes 0-15, 1: lanes 16-31
    const int l15     = lane & 15;
    const int colBase = wave * 16;

    v8f acc = {};
#pragma unroll 4
    for (int k = 0; k < KDIM; k += 4) {
        const int ka = k + (half << 1);          // +0 (lanes 0-15) or +2 (lanes 16-31)
        v2f a, b;
        a.x = As[l15][ka];
        a.y = As[l15][ka + 1];
        b.x = W[(size_t)ka * NDIM + colBase + l15];
        b.y = W[(size_t)(ka + 1) * NDIM + colBase + l15];
        acc = __builtin_amdgcn_wmma_f32_16x16x4_f32(
            /*neg_a=*/false, a, /*neg_b=*/false, b,
            /*c_mod=*/(short)0, acc, /*reuse_a=*/false, /*reuse_b=*/false);
    }

#pragma unroll
    for (int v = 0; v < 8; ++v) {
        const int row = row0 + v + half * 8;
        H[(size_t)row * NDIM + colBase + l15] = acc[v] * norm[row];
    }
}

// ---------------------------------------------------------------------------
// Edge scatter: agg[dst[e]][c] += h[src[e]][c].
// gridDim.x = E, blockDim.x = C. Coalesced gather + coalesced hw float atomics.
// ---------------------------------------------------------------------------
template <int C>
__global__ void scatter_kernel(const float* __restrict__ h, const int* __restrict__ src,
                               const int* __restrict__ dst, float* __restrict__ agg) {
    const int e = blockIdx.x;
    const int c = threadIdx.x;
    const int s = src[e];
    const int d = dst[e];
    const float v = h[(size_t)s * C + c];
    unsafeAtomicAdd(&agg[(size_t)d * C + c], v);   // global_atomic_add_f32, no CAS loop
}

// ---------------------------------------------------------------------------
// out[i][c] = relu((agg[i][c] + h[i][c]) * norm_dst[i] + bias[c]), in place on agg.
// h[i] supplies the self-loop message (already scaled by norm_src[i]).
// ---------------------------------------------------------------------------
template <int C>
__global__ void finalize_kernel(float* __restrict__ agg, const float* __restrict__ h,
                                const float* __restrict__ norm_dst,
                                const float* __restrict__ bias) {
    const int i = blockIdx.x;
    const int c = threadIdx.x;
    const size_t idx = (size_t)i * C + c;
    float v = (agg[idx] + h[idx]) * norm_dst[i] + bias[c];
    agg[idx] = v > 0.0f ? v : 0.0f;
}

// ---------------------------------------------------------------------------
// Host-side launch.
// ---------------------------------------------------------------------------
extern "C" void kernel_launch(void* const* d_in, const int* in_sizes, int n_in,
                              void* d_out, int out_size, void* d_ws, size_t ws_size,
                              hipStream_t stream) {
    const float* feats = (const float*)d_in[0];
    const int*   src   = (const int*)d_in[1];
    const int*   dst   = (const int*)d_in[2];
    const float* W0    = (const float*)d_in[3];
    const float* b0    = (const float*)d_in[4];
    const float* W1    = (const float*)d_in[5];
    const float* b1    = (const float*)d_in[6];
    float*       out   = (float*)d_out;

    const int N = in_sizes[0] / IN_CH;   // 50000 (divisible by 16)
    const int E = in_sizes[1];           // 800000

    // Workspace carve-out (256B aligned slices).
    char*  ws  = (char*)d_ws;
    size_t off = 0;
    auto carve = [&](size_t bytes) -> char* {
        char* p = ws + off;
        off += (bytes + 255) & ~(size_t)255;
        return p;
    };
    float* norm_src = (float*)carve((size_t)N * sizeof(float));
    float* norm_dst = (float*)carve((size_t)N * sizeof(float));
    float* h        = (float*)carve((size_t)N * HID_CH * sizeof(float)); // h0; reused for h1
    float* agg      = (float*)carve((size_t)N * HID_CH * sizeof(float)); // agg0 -> x1

    // Zero accumulators & counters (d_out / d_ws are poisoned by the harness).
    hipMemsetAsync(norm_src, 0, (size_t)N * sizeof(float), stream);
    hipMemsetAsync(norm_dst, 0, (size_t)N * sizeof(float), stream);
    hipMemsetAsync(agg,      0, (size_t)N * HID_CH * sizeof(float), stream);
    hipMemsetAsync(out,      0, (size_t)N * OUT_CH * sizeof(float), stream);

    // Degrees -> norms (self-loop folded in as +1).
    degree_kernel<<<(E + 255) / 256, 256, 0, stream>>>(src, dst, (int*)norm_src, (int*)norm_dst, E);
    norm_kernel<<<(N + 255) / 256, 256, 0, stream>>>(norm_src, N);
    norm_kernel<<<(N + 255) / 256, 256, 0, stream>>>(norm_dst, N);

    // ---- Layer 0: h0 = (feats @ W0) * norm_src ; agg0 = scatter ; x1 = relu(...) ----
    gemm_norm_kernel<IN_CH, HID_CH><<<N / 16, (HID_CH / 16) * 32, 0, stream>>>(feats, W0, norm_src, h);
    scatter_kernel<HID_CH><<<E, HID_CH, 0, stream>>>(h, src, dst, agg);
    finalize_kernel<HID_CH><<<N, HID_CH, 0, stream>>>(agg, h, norm_dst, b0);   // agg now = x1

    // ---- Layer 1: h1 = (x1 @ W1) * norm_src ; scatter into d_out ; relu in place ----
    gemm_norm_kernel<HID_CH, OUT_CH><<<N / 16, (OUT_CH / 16) * 32, 0, stream>>>(agg, W1, norm_src, h);
    scatter_kernel<OUT_CH><<<E, OUT_CH, 0, stream>>>(h, src, dst, out);
    finalize_kernel<OUT_CH><<<N, OUT_CH, 0, stream>>>(out, h, norm_dst, b1);
}